// GreenBlock_50448685859377
// MI455X (gfx1250) — compile-verified
//
#include <hip/hip_runtime.h>
#include <hip/hip_bf16.h>

typedef float v2f __attribute__((ext_vector_type(2)));
typedef float v8f __attribute__((ext_vector_type(8)));

#define C_DIM 64
#define BATCH 4

// ---------------------------------------------------------------------------
// Generic fp32 WMMA GEMM:  C[m, coff+n] = act( A[m,:] . W[n,:] )   (out = A @ W^T)
// A: [M, K] dense (row stride K). W: [NOUT, K]. C row stride = ldc.
// One wave per 16-row tile, 4 waves (128 threads) per block, M % 64 == 0.
// V_WMMA_F32_16X16X4_F32: A frag 16x4 (v2f/lane), B frag 4x16 (v2f/lane),
// acc 16x16 f32 (v8f/lane).
// All B fragments for a column tile are preloaded into registers so the
// WMMA chain is not serialized behind per-fragment load latency; the n0
// loop is fully unrolled so loads for the next tile pipeline under the
// current tile's WMMAs.
// ---------------------------------------------------------------------------
template <int K, int NOUT, bool RELU>
__global__ __launch_bounds__(128)
void gemm_wmma_f32(const float* __restrict__ A, const float* __restrict__ W,
                   float* __restrict__ Cmat, int ldc, int coff)
{
    const int lane = threadIdx.x & 31;
    const int wave = threadIdx.x >> 5;
    const int lm   = lane & 15;   // M (A/D) or N (B/D) index within tile
    const int lh   = lane >> 4;   // half-wave selector
    const long row0 = ((long)blockIdx.x * 4 + wave) * 16;

    // Preload all A fragments for this 16-row tile (K/4 chunks).
    v2f afrag[K / 4];
    const float* arow = A + (row0 + lm) * (long)K;
#pragma unroll
    for (int kc = 0; kc < K / 4; ++kc)
        afrag[kc] = *(const v2f*)(arow + kc * 4 + 2 * lh);

#pragma unroll
    for (int n0 = 0; n0 < NOUT; n0 += 16) {
        const float* wrow = W + (n0 + lm) * (long)K;   // B[k,n] = W[n,k]

        // Stage the whole column tile's B fragments in registers first.
        v2f bfrag[K / 4];
#pragma unroll
        for (int kc = 0; kc < K / 4; ++kc)
            bfrag[kc] = *(const v2f*)(wrow + kc * 4 + 2 * lh);

        v8f acc = {0.f, 0.f, 0.f, 0.f, 0.f, 0.f, 0.f, 0.f};
#pragma unroll
        for (int kc = 0; kc < K / 4; ++kc)
            acc = __builtin_amdgcn_wmma_f32_16x16x4_f32(
                false, afrag[kc], false, bfrag[kc], (short)0, acc, false, false);

        // D element (M = v + 8*lh, N = lm)
        float* crow = Cmat + (row0 + 8 * lh) * (long)ldc + coff + n0 + lm;
#pragma unroll
        for (int v = 0; v < 8; ++v) {
            float val = acc[v];
            if (RELU) val = fmaxf(val, 0.f);
            crow[(long)v * ldc] = val;
        }
    }
}

// ---------------------------------------------------------------------------
// Degree of each target node (self-loops added analytically later).
// ---------------------------------------------------------------------------
__global__ void degree_kernel(const int* __restrict__ col, float* __restrict__ deg, int E)
{
    int i = blockIdx.x * 256 + threadIdx.x;
    if (i < E) atomicAdd(&deg[col[i]], 1.0f);
}

__global__ void invsqrt_kernel(float* __restrict__ deg, int N)
{
    int i = blockIdx.x * 256 + threadIdx.x;
    if (i < N) deg[i] = rsqrtf(deg[i] + 1.0f);   // +1 = self loop; always > 0
}

// ---------------------------------------------------------------------------
// Edge gather/scale/scatter-add: one 32-thread wave per edge, 2 channels/lane,
// all 4 batches per lane. agg[b,t,c] += h[b,r,c] * dis[r]*dis[t].
// ---------------------------------------------------------------------------
__global__ __launch_bounds__(256)
void scatter_kernel(const int* __restrict__ row, const int* __restrict__ col,
                    const float* __restrict__ h, const float* __restrict__ dis,
                    float* __restrict__ agg, int E, int N)
{
    int idx = blockIdx.x * 256 + threadIdx.x;
    int e = idx >> 5;
    if (e >= E) return;
    int c = (idx & 31) * 2;
    int r = row[e];
    int t = col[e];
    float w = dis[r] * dis[t];
#pragma unroll
    for (int b = 0; b < BATCH; ++b) {
        const float2 hv = *(const float2*)(h + ((size_t)b * N + r) * C_DIM + c);
        float* dst = agg + ((size_t)b * N + t) * C_DIM + c;
        atomicAdd(dst,     hv.x * w);
        atomicAdd(dst + 1, hv.y * w);
    }
}

// ---------------------------------------------------------------------------
// fst = sigmoid(agg + h*dis^2 + bias)   (self-loop contribution folded in here)
// ---------------------------------------------------------------------------
__global__ void finalize_kernel(const float* __restrict__ h, const float* __restrict__ dis,
                                const float* __restrict__ bias, float* __restrict__ agg,
                                int N, int total)
{
    int i = blockIdx.x * 256 + threadIdx.x;
    if (i >= total) return;
    int c  = i & (C_DIM - 1);
    int rg = i >> 6;          // global row (b*N + n)
    int n  = rg % N;
    float d = dis[n];
    float v = agg[i] + h[i] * d * d + bias[c];
    agg[i] = 1.0f / (1.0f + expf(-v));
}

// ---------------------------------------------------------------------------
extern "C" void kernel_launch(void* const* d_in, const int* in_sizes, int n_in,
                              void* d_out, int out_size, void* d_ws, size_t ws_size,
                              hipStream_t stream)
{
    const float* x      = (const float*)d_in[0];
    const int*   eidx   = (const int*)d_in[1];   // [2, E] (row; col)
    const float* lin_w  = (const float*)d_in[2];
    const float* bias   = (const float*)d_in[3];
    const float* up1_w  = (const float*)d_in[4];
    const float* up2_w  = (const float*)d_in[5];
    const float* lo1_w  = (const float*)d_in[6];
    const float* lo2_w  = (const float*)d_in[7];
    const float* last_w = (const float*)d_in[8];
    float* out = (float*)d_out;

    const int E  = in_sizes[1] / 2;
    const int BN = in_sizes[0] / C_DIM;   // B*N = 80000
    const int N  = BN / BATCH;            // 20000

    const int* row = eidx;
    const int* col = eidx + E;

    // Workspace layout (floats): h | agg/fst | combined | dis
    float* h    = (float*)d_ws;                          // BN*64  (later: up/lo temp)
    float* fst  = h    + (size_t)BN * C_DIM;             // BN*64  (agg, then sigmoid)
    float* comb = fst  + (size_t)BN * C_DIM;             // BN*128
    float* dis  = comb + (size_t)BN * 2 * C_DIM;         // N      (deg, then deg^-1/2)

    // Atomically-accumulated buffers must be zeroed every call.
    hipMemsetAsync(fst, 0, (size_t)BN * C_DIM * sizeof(float), stream);
    hipMemsetAsync(dis, 0, (size_t)N * sizeof(float), stream);

    const int gemm_blocks = BN / 64;   // 1250; 4 waves x 16 rows per block

    // 1) h = x @ lin_w^T
    gemm_wmma_f32<64, 64, false><<<gemm_blocks, 128, 0, stream>>>(x, lin_w, h, 64, 0);

    // 2) degree -> dis = (deg+1)^-1/2
    degree_kernel <<<(E + 255) / 256, 256, 0, stream>>>(col, dis, E);
    invsqrt_kernel<<<(N + 255) / 256, 256, 0, stream>>>(dis, N);

    // 3) gather/scale/scatter-add over real edges (self loops folded into finalize)
    scatter_kernel<<<(E * 32 + 255) / 256, 256, 0, stream>>>(row, col, h, dis, fst, E, N);

    // 4) fst = sigmoid(agg + h*dis^2 + bias)
    finalize_kernel<<<(BN * C_DIM + 255) / 256, 256, 0, stream>>>(h, dis, bias, fst, N, BN * C_DIM);

    // 5) upper = (fst @ up1^T) @ up2^T -> combined[:, 0:64]
    gemm_wmma_f32<64, 64, false><<<gemm_blocks, 128, 0, stream>>>(fst, up1_w, h, 64, 0);
    gemm_wmma_f32<64, 64, false><<<gemm_blocks, 128, 0, stream>>>(h, up2_w, comb, 128, 0);

    // 6) lower = (fst @ lo1^T) @ lo2^T -> combined[:, 64:128]
    gemm_wmma_f32<64, 64, false><<<gemm_blocks, 128, 0, stream>>>(fst, lo1_w, h, 64, 0);
    gemm_wmma_f32<64, 64, false><<<gemm_blocks, 128, 0, stream>>>(h, lo2_w, comb, 128, 64);

    // 7) out = relu(combined @ last_w^T)
    gemm_wmma_f32<128, 128, true><<<gemm_blocks, 128, 0, stream>>>(comb, last_w, out, 128, 0);
}